// GNN_node_Virtualnode_90915867722227
// MI455X (gfx1250) — compile-verified
//
#include <hip/hip_runtime.h>

#define NN 50000
#define EE 640000
#define DD 128
#define LL 5
#define GG 128
#define EPB 64   // edges per block in the fused edge kernel
#define NPB 64   // nodes per block in the pooling kernel

typedef float v2f __attribute__((ext_vector_type(2)));
typedef float v8f __attribute__((ext_vector_type(8)));

// ---------------------------------------------------------------------------
// fp32 GEMM with bias via V_WMMA_F32_16X16X4_F32.
// C[M x Ncol] = A[M x K] @ B[K x Ncol] + bias.
// M mult of 16, K mult of 4, Ncol mult of 64.
// One wave computes a 16x64 strip: 4 accumulators share one A fragment per
// k-step (4x less A traffic, 4 independent WMMAs in flight per iteration).
// A frag (16x4): lane = (khalf?16:0)+m, v[j] = A[m][2*khalf+j]
// B frag (4x16): lane = (khalf?16:0)+n, v[j] = B[2*khalf+j][n]
// C/D (16x16):   vgpr g, lane = (mhalf?16:0)+n, row = g + 8*mhalf
// ---------------------------------------------------------------------------
__global__ void __launch_bounds__(256)
wmma_gemm_bias_k(const float* __restrict__ A, const float* __restrict__ B,
                 const float* __restrict__ bias, float* __restrict__ C,
                 int M, int K, int Ncol) {
  const int lane = threadIdx.x & 31;
  const int wave = threadIdx.x >> 5;
  const int stripsN = Ncol >> 6;                   // 64-wide column strips
  const int total = (M >> 4) * stripsN;
  const int id = blockIdx.x * 8 + wave;            // wave-uniform
  if (id >= total) return;                         // whole wave exits or stays
  const int tm = id / stripsN;
  const int ts = id - tm * stripsN;
  const int m0 = tm << 4;
  const int n0 = ts << 6;
  const int lm = lane & 15;
  const int hi = lane >> 4;                        // 0/1: k-half (A,B), m-half (C)

  v8f acc[4] = {v8f{}, v8f{}, v8f{}, v8f{}};
  const float* __restrict__ Arow = A + (size_t)(m0 + lm) * K + 2 * hi;
  const float* __restrict__ Bc   = B + n0 + lm;

  for (int k0 = 0; k0 < K; k0 += 4) {
    const v2f a = *reinterpret_cast<const v2f*>(Arow + k0);   // 8B-aligned
    const float* __restrict__ Br0 = Bc + (size_t)(k0 + 2 * hi) * Ncol;
    const float* __restrict__ Br1 = Br0 + Ncol;
    v2f b0, b1, b2, b3;
    b0.x = Br0[0];  b0.y = Br1[0];
    b1.x = Br0[16]; b1.y = Br1[16];
    b2.x = Br0[32]; b2.y = Br1[32];
    b3.x = Br0[48]; b3.y = Br1[48];
    acc[0] = __builtin_amdgcn_wmma_f32_16x16x4_f32(false, a, false, b0, (short)0, acc[0], false, false);
    acc[1] = __builtin_amdgcn_wmma_f32_16x16x4_f32(false, a, false, b1, (short)0, acc[1], false, false);
    acc[2] = __builtin_amdgcn_wmma_f32_16x16x4_f32(false, a, false, b2, (short)0, acc[2], false, false);
    acc[3] = __builtin_amdgcn_wmma_f32_16x16x4_f32(false, a, false, b3, (short)0, acc[3], false, false);
  }

#pragma unroll
  for (int t = 0; t < 4; ++t) {
    const int col = n0 + 16 * t + lm;
    const float bv = bias[col];
#pragma unroll
    for (int g = 0; g < 8; ++g) {
      const int row = m0 + g + 8 * hi;
      C[(size_t)row * Ncol + col] = acc[t][g] + bv;
    }
  }
}

// ---------------------------------------------------------------------------
// Elementwise / graph kernels
// ---------------------------------------------------------------------------
__global__ void __launch_bounds__(256) fill_zero_k(float* __restrict__ p, size_t n) {
  size_t i = (size_t)blockIdx.x * blockDim.x + threadIdx.x;
  const size_t stride = (size_t)gridDim.x * blockDim.x;
  for (; i < n; i += stride) p[i] = 0.0f;
}

__global__ void __launch_bounds__(128) gather_emb_k(const int* __restrict__ x,
                                                    const float* __restrict__ emb,
                                                    float* __restrict__ h) {
  const int i = blockIdx.x, j = threadIdx.x;
  h[(size_t)i * DD + j] = emb[(size_t)x[i] * DD + j];
}

__global__ void __launch_bounds__(128) bcast_vn_k(const float* __restrict__ vn_emb,
                                                  float* __restrict__ vn) {
  vn[(size_t)blockIdx.x * DD + threadIdx.x] = vn_emb[threadIdx.x];
}

__global__ void __launch_bounds__(128) add_vn_k(const float* __restrict__ h,
                                                const float* __restrict__ vn,
                                                const int* __restrict__ batch,
                                                float* __restrict__ h_in) {
  const int i = blockIdx.x, j = threadIdx.x;
  h_in[(size_t)i * DD + j] = h[(size_t)i * DD + j] + vn[(size_t)batch[i] * DD + j];
}

// Fused edge encoder + message + relu + segment-sum scatter.
// Block = 128 threads, EPB edges. Weight panel + bias + this block's edge
// metadata staged in LDS once; barrier-free edge loop afterwards with a
// prefetch of the next edge's gather row.
__global__ void __launch_bounds__(128) edge_msg_agg_k(
    const float* __restrict__ ea, const float* __restrict__ eW,
    const float* __restrict__ eb, const float* __restrict__ h_in,
    const int* __restrict__ src, const int* __restrict__ dst,
    float* __restrict__ agg) {
  __shared__ float sW[7 * DD];
  __shared__ float sb[DD];
  __shared__ int   ssrc[EPB];
  __shared__ int   sdst[EPB];
  __shared__ float sattr[EPB * 7];

  const int j = threadIdx.x;
  const int e0 = blockIdx.x * EPB;

#pragma unroll
  for (int t = j; t < 7 * DD; t += 128) sW[t] = eW[t];
  sb[j] = eb[j];
  if (j < EPB)            ssrc[j]        = src[e0 + j];
  else                    sdst[j - EPB]  = dst[e0 + j - EPB];
  for (int t = j; t < EPB * 7; t += 128) sattr[t] = ea[(size_t)e0 * 7 + t];
  __syncthreads();

  __builtin_prefetch(h_in + (size_t)ssrc[0] * DD + j, 0, 3);   // global_prefetch_b8

  for (int k = 0; k < EPB; ++k) {
    if (k + 1 < EPB)
      __builtin_prefetch(h_in + (size_t)ssrc[k + 1] * DD + j, 0, 3);
    const float* __restrict__ hrow = h_in + (size_t)ssrc[k] * DD;
    const float* __restrict__ at = &sattr[k * 7];
    float acc = sb[j];
#pragma unroll
    for (int q = 0; q < 7; ++q) acc = fmaf(at[q], sW[q * DD + j], acc);
    float m = hrow[j] + acc;
    m = m > 0.0f ? m : 0.0f;
    atomicAdd(agg + (size_t)sdst[k] * DD + j, m);    // global_atomic_add_f32
  }
}

// agg = (1+eps[l])*h_in + agg
__global__ void __launch_bounds__(256) gin_combine_k(const float* __restrict__ h_in,
                                                     float* __restrict__ agg,
                                                     const float* __restrict__ eps,
                                                     int l, size_t n) {
  const float c = 1.0f + eps[l];
  size_t i = (size_t)blockIdx.x * blockDim.x + threadIdx.x;
  const size_t stride = (size_t)gridDim.x * blockDim.x;
  for (; i < n; i += stride) agg[i] = c * h_in[i] + agg[i];
}

// Column sums / sum-of-squares (blockDim.x == cols); coalesced row sweep.
__global__ void bn_stats_k(const float* __restrict__ x, int rows, int cols,
                           float* __restrict__ sums, float* __restrict__ sumsq) {
  const int c = threadIdx.x;
  float s = 0.0f, ss = 0.0f;
  for (int r = blockIdx.x; r < rows; r += gridDim.x) {
    const float v = x[(size_t)r * cols + c];
    s += v;
    ss += v * v;
  }
  atomicAdd(&sums[c], s);
  atomicAdd(&sumsq[c], ss);
}

__global__ void __launch_bounds__(256) bn_apply_k(
    const float* __restrict__ xin, float* __restrict__ xout,
    const float* __restrict__ sums, const float* __restrict__ sumsq,
    const float* __restrict__ gma, const float* __restrict__ bta,
    int rows, int cols, int do_relu) {
  const size_t n = (size_t)rows * cols;
  const float inv = 1.0f / (float)rows;
  size_t i = (size_t)blockIdx.x * blockDim.x + threadIdx.x;
  const size_t stride = (size_t)gridDim.x * blockDim.x;
  for (; i < n; i += stride) {
    const int c = (int)(i % cols);
    const float m = sums[c] * inv;
    const float var = sumsq[c] * inv - m * m;
    float y = (xin[i] - m) * rsqrtf(var + 1e-5f) * gma[c] + bta[c];
    if (do_relu) y = y > 0.0f ? y : 0.0f;
    xout[i] = y;
  }
}

// Graph pooling: batch is sorted, so accumulate runs locally and flush one
// atomic per (run, column) instead of per (node, column).
__global__ void __launch_bounds__(128) pool_k(const float* __restrict__ h_in,
                                              const int* __restrict__ batch,
                                              float* __restrict__ pooled) {
  const int j = threadIdx.x;
  const int i0 = blockIdx.x * NPB;
  const int i1 = (i0 + NPB < NN) ? (i0 + NPB) : NN;
  int cur = batch[i0];
  float acc = 0.0f;
  for (int i = i0; i < i1; ++i) {
    const int g = batch[i];
    if (g != cur) {
      atomicAdd(pooled + (size_t)cur * DD + j, acc);
      acc = 0.0f;
      cur = g;
    }
    acc += h_in[(size_t)i * DD + j];
  }
  atomicAdd(pooled + (size_t)cur * DD + j, acc);
}

__global__ void __launch_bounds__(256) add2_k(const float* __restrict__ a,
                                              const float* __restrict__ b,
                                              float* __restrict__ o, size_t n) {
  size_t i = (size_t)blockIdx.x * blockDim.x + threadIdx.x;
  const size_t stride = (size_t)gridDim.x * blockDim.x;
  for (; i < n; i += stride) o[i] = a[i] + b[i];
}

// ---------------------------------------------------------------------------
// Host orchestration
// ---------------------------------------------------------------------------
static inline int gemm_blocks(int M, int Ncol) {
  const int waves = (M / 16) * (Ncol / 64);
  return (waves + 7) / 8;
}

extern "C" void kernel_launch(void* const* d_in, const int* in_sizes, int n_in,
                              void* d_out, int out_size, void* d_ws, size_t ws_size,
                              hipStream_t stream) {
  const int*   x        = (const int*)d_in[0];
  const int*   eidx     = (const int*)d_in[1];
  const float* eattr    = (const float*)d_in[2];
  const int*   batch    = (const int*)d_in[3];
  const float* node_emb = (const float*)d_in[4];
  const float* vn_emb   = (const float*)d_in[5];
  const float* eps      = (const float*)d_in[6];
  const float* edge_W   = (const float*)d_in[7];
  const float* edge_b   = (const float*)d_in[8];
  const float* W1       = (const float*)d_in[9];
  const float* b1       = (const float*)d_in[10];
  const float* bn1_g    = (const float*)d_in[11];
  const float* bn1_b    = (const float*)d_in[12];
  const float* W2       = (const float*)d_in[13];
  const float* b2       = (const float*)d_in[14];
  const float* bn_g     = (const float*)d_in[15];
  const float* bn_b     = (const float*)d_in[16];
  const float* vn_W1    = (const float*)d_in[17];
  const float* vn_b1    = (const float*)d_in[18];
  const float* vn_bn1_g = (const float*)d_in[19];
  const float* vn_bn1_b = (const float*)d_in[20];
  const float* vn_W2    = (const float*)d_in[21];
  const float* vn_b2    = (const float*)d_in[22];
  const float* vn_bn2_g = (const float*)d_in[23];
  const float* vn_bn2_b = (const float*)d_in[24];

  const int* src = eidx;
  const int* dst = eidx + EE;

  float* ws = (float*)d_ws;
  const size_t ND = (size_t)NN * DD;
  const size_t GD = (size_t)GG * DD;
  float* h     = ws;                // N x D
  float* h_in  = h + ND;            // N x D
  float* agg   = h_in + ND;         // N x D  (also holds z)
  float* z2    = agg + ND;          // N x 2D
  float* vn    = z2 + 2 * ND;       // G x D
  float* vtmp  = vn + GD;           // G x D
  float* v2    = vtmp + GD;         // G x 2D
  float* sums  = v2 + 2 * GD;       // 256
  float* sumsq = sums + 256;        // 256
  float* outp  = (float*)d_out;

  gather_emb_k<<<NN, 128, 0, stream>>>(x, node_emb, h);
  bcast_vn_k<<<GG, 128, 0, stream>>>(vn_emb, vn);

  for (int l = 0; l < LL; ++l) {
    // h_in = h + vn[batch]
    add_vn_k<<<NN, 128, 0, stream>>>(h, vn, batch, h_in);

    // agg = segment_sum(relu(h_in[src] + edge_attr @ eW + eb), dst)
    fill_zero_k<<<2048, 256, 0, stream>>>(agg, ND);
    edge_msg_agg_k<<<EE / EPB, 128, 0, stream>>>(eattr,
        edge_W + (size_t)l * 7 * DD, edge_b + (size_t)l * DD,
        h_in, src, dst, agg);

    // z = (1+eps)*h_in + agg  (in place into agg)
    gin_combine_k<<<2048, 256, 0, stream>>>(h_in, agg, eps, l, ND);

    // z2 = relu(BN(z @ W1 + b1))
    wmma_gemm_bias_k<<<gemm_blocks(NN, 2 * DD), 256, 0, stream>>>(
        agg, W1 + (size_t)l * DD * 2 * DD, b1 + (size_t)l * 2 * DD,
        z2, NN, DD, 2 * DD);
    fill_zero_k<<<2, 256, 0, stream>>>(sums, 512);
    bn_stats_k<<<512, 2 * DD, 0, stream>>>(z2, NN, 2 * DD, sums, sumsq);
    bn_apply_k<<<2048, 256, 0, stream>>>(z2, z2, sums, sumsq,
        bn1_g + (size_t)l * 2 * DD, bn1_b + (size_t)l * 2 * DD, NN, 2 * DD, 1);

    // h = [relu](BN(z2 @ W2 + b2));  last layer: no relu, write d_out
    wmma_gemm_bias_k<<<gemm_blocks(NN, DD), 256, 0, stream>>>(
        z2, W2 + (size_t)l * 2 * DD * DD, b2 + (size_t)l * DD,
        h, NN, 2 * DD, DD);
    fill_zero_k<<<2, 256, 0, stream>>>(sums, 512);
    bn_stats_k<<<512, DD, 0, stream>>>(h, NN, DD, sums, sumsq);
    {
      float* target = (l == LL - 1) ? outp : h;
      bn_apply_k<<<2048, 256, 0, stream>>>(h, target, sums, sumsq,
          bn_g + (size_t)l * DD, bn_b + (size_t)l * DD, NN, DD,
          (l == LL - 1) ? 0 : 1);
    }

    // Virtual-node update
    if (l < LL - 1) {
      fill_zero_k<<<64, 256, 0, stream>>>(vtmp, GD);
      pool_k<<<(NN + NPB - 1) / NPB, 128, 0, stream>>>(h_in, batch, vtmp);
      add2_k<<<64, 256, 0, stream>>>(vtmp, vn, vtmp, GD);

      wmma_gemm_bias_k<<<gemm_blocks(GG, 2 * DD), 256, 0, stream>>>(
          vtmp, vn_W1 + (size_t)l * DD * 2 * DD, vn_b1 + (size_t)l * 2 * DD,
          v2, GG, DD, 2 * DD);
      fill_zero_k<<<2, 256, 0, stream>>>(sums, 512);
      bn_stats_k<<<128, 2 * DD, 0, stream>>>(v2, GG, 2 * DD, sums, sumsq);
      bn_apply_k<<<128, 256, 0, stream>>>(v2, v2, sums, sumsq,
          vn_bn1_g + (size_t)l * 2 * DD, vn_bn1_b + (size_t)l * 2 * DD, GG, 2 * DD, 1);

      wmma_gemm_bias_k<<<gemm_blocks(GG, DD), 256, 0, stream>>>(
          v2, vn_W2 + (size_t)l * 2 * DD * DD, vn_b2 + (size_t)l * DD,
          vn, GG, 2 * DD, DD);
      fill_zero_k<<<2, 256, 0, stream>>>(sums, 512);
      bn_stats_k<<<128, DD, 0, stream>>>(vn, GG, DD, sums, sumsq);
      bn_apply_k<<<64, 256, 0, stream>>>(vn, vn, sums, sumsq,
          vn_bn2_g + (size_t)l * DD, vn_bn2_b + (size_t)l * DD, GG, DD, 1);
    }
  }
}